// SelfAttnPooling_36103495090826
// MI455X (gfx1250) — compile-verified
//
#include <hip/hip_runtime.h>
#include <float.h>
#include <stdint.h>

// MI455X / gfx1250 single-pass segment-softmax pooling.
// x is read exactly once from HBM (1.02 GB -> ~44us floor @ 23.3 TB/s).
// One wave32 per segment, online softmax, WMMA f32 16x16x4 for scores,
// async global->LDS double-buffered tile streaming.

typedef __attribute__((ext_vector_type(2))) float v2f;
typedef __attribute__((ext_vector_type(8))) float v8f;
typedef int v4i_vs __attribute__((vector_size(16)));   // matches builtin param type

#define HID  128
#define TILE 16

// ---- async global -> LDS (ASYNCcnt path) --------------------------------
__device__ __forceinline__ void async_load_16B(uint32_t lds_byte, const float* g) {
#if __has_builtin(__builtin_amdgcn_global_load_async_to_lds_b128)
  __builtin_amdgcn_global_load_async_to_lds_b128(
      (__attribute__((address_space(1))) v4i_vs*)(uintptr_t)g,
      (__attribute__((address_space(3))) v4i_vs*)(uintptr_t)(uint32_t)lds_byte,
      0, 0);
#else
  asm volatile("global_load_async_to_lds_b128 %0, %1, off"
               :: "v"(lds_byte), "v"((unsigned long long)(uintptr_t)g)
               : "memory");
#endif
}

__device__ __forceinline__ void wait_asynccnt_0() {
  asm volatile("" ::: "memory");
#if __has_builtin(__builtin_amdgcn_s_wait_asynccnt)
  __builtin_amdgcn_s_wait_asynccnt(0);
#else
  asm volatile("s_wait_asynccnt 0" ::: "memory");
#endif
  asm volatile("" ::: "memory");
}

__device__ __forceinline__ void wait_asynccnt_16() {
  asm volatile("" ::: "memory");
#if __has_builtin(__builtin_amdgcn_s_wait_asynccnt)
  __builtin_amdgcn_s_wait_asynccnt(16);
#else
  asm volatile("s_wait_asynccnt 16" ::: "memory");
#endif
  asm volatile("" ::: "memory");
}

__device__ __forceinline__ void wait_dscnt_0() {
#if __has_builtin(__builtin_amdgcn_s_wait_dscnt)
  __builtin_amdgcn_s_wait_dscnt(0);
#else
  asm volatile("s_wait_dscnt 0" ::: "memory");
#endif
  asm volatile("" ::: "memory");
}

__global__ __launch_bounds__(32)
void attn_pool_kernel(const float* __restrict__ x, const float* __restrict__ w,
                      const int* __restrict__ seg, float* __restrict__ out,
                      int n_nodes) {
  // LDS: [0,128) = w ; then two 16x128 f32 tile buffers.
  __shared__ float smem[HID + 2 * TILE * HID];
  const int g    = (int)blockIdx.x;
  const int lane = (int)threadIdx.x;

  // Stage w into LDS (one wave -> DS ops are wave-ordered; barrier for safety).
  *(float4*)(smem + lane * 4) = *(const float4*)(w + lane * 4);
  __syncthreads();

  // Segment range via binary search (segment_ids sorted ascending).
  int lo = 0, hi = n_nodes;
  while (lo < hi) { int mid = (lo + hi) >> 1; if (seg[mid] < g)  lo = mid + 1; else hi = mid; }
  const int sBeg = lo;
  hi = n_nodes;
  while (lo < hi) { int mid = (lo + hi) >> 1; if (seg[mid] <= g) lo = mid + 1; else hi = mid; }
  const int sEnd = lo;
  const int cnt  = sEnd - sBeg;

  float m = -FLT_MAX, den = 0.f;
  float ax = 0.f, ay = 0.f, az = 0.f, aw = 0.f;

  const uint32_t ldsTile0 = (uint32_t)(uintptr_t)(smem + HID);
  const int laneB = lane * 16;  // 16 bytes per lane per row

  if (cnt > 0) {
    const int ntiles = (cnt + TILE - 1) / TILE;

    // Issue one 16-row tile (rows past segment end clamped to a valid row;
    // they are never consumed, keeps the wait-count uniform at 16).
    auto issue = [&](int t, int buf) {
      const int base = sBeg + t * TILE;
      const uint32_t l0 = ldsTile0 + (uint32_t)buf * (TILE * HID * 4) + (uint32_t)laneB;
#pragma unroll
      for (int j = 0; j < TILE; ++j) {
        int node = base + j;
        if (node >= sEnd) node = sEnd - 1;
        async_load_16B(l0 + (uint32_t)j * (HID * 4),
                       x + (size_t)node * HID + lane * 4);
      }
    };

    issue(0, 0);
    for (int t = 0; t < ntiles; ++t) {
      const int  buf     = t & 1;
      const bool hasNext = (t + 1) < ntiles;

      wait_dscnt_0();                    // prior reads of buf^1 done before overwrite
      if (hasNext) issue(t + 1, buf ^ 1);
      if (hasNext) wait_asynccnt_16();   // exactly the 16 next-tile loads may remain
      else         wait_asynccnt_0();

      const float* tile = smem + HID + buf * (TILE * HID);

      // ---- 16 scores via chained V_WMMA_F32_16X16X4_F32 over K=128 ----
      // A layout (16x4 f32): lanes 0-15 row=lane hold K=0,1; lanes 16-31 hold K=2,3.
      // B broadcasts the same w K-chunk to all 16 N columns -> every D column == scores.
      const int ksel = (lane >= 16) ? 2 : 0;
      const float* aRow = tile + (lane & 15) * HID + ksel;
      const float* bRow = smem + ksel;
      v8f c = {0.f, 0.f, 0.f, 0.f, 0.f, 0.f, 0.f, 0.f};
#pragma unroll
      for (int kc = 0; kc < HID / 4; ++kc) {
        v2f a = *(const v2f*)(aRow + kc * 4);   // ds_load_b64
        v2f b = *(const v2f*)(bRow + kc * 4);   // ds_load_b64
        c = __builtin_amdgcn_wmma_f32_16x16x4_f32(false, a, false, b,
                                                  (short)0, c, false, false);
      }
      // D layout: VGPR j -> M=j (lanes 0-15), M=j+8 (lanes 16-31).
      float sc[16];
#pragma unroll
      for (int j = 0; j < 8; ++j) {
        sc[j]     = __shfl(c[j], 0, 32);
        sc[j + 8] = __shfl(c[j], 16, 32);
      }

      int r = sEnd - (sBeg + t * TILE);
      if (r > TILE) r = TILE;

      float tmax = -FLT_MAX;
#pragma unroll
      for (int j = 0; j < TILE; ++j)
        if (j < r) tmax = fmaxf(tmax, sc[j]);

      const float mNew  = fmaxf(m, tmax);
      const float alpha = __expf(m - mNew);   // first tile: exp(-huge) == 0
      den *= alpha; ax *= alpha; ay *= alpha; az *= alpha; aw *= alpha;

#pragma unroll
      for (int j = 0; j < TILE; ++j) {
        if (j < r) {  // uniform branch; rows >= r never touched
          const float e = __expf(sc[j] - mNew);
          den += e;
          const float4 xv = *(const float4*)(tile + j * HID + lane * 4); // ds_load_b128
          ax = fmaf(e, xv.x, ax); ay = fmaf(e, xv.y, ay);
          az = fmaf(e, xv.z, az); aw = fmaf(e, xv.w, aw);
        }
      }
      m = mNew;
    }
  }

  float4 o;
  if (cnt > 0) {
    const float inv = 1.0f / den;          // den >= 1 for non-empty segments
    o.x = ax * inv; o.y = ay * inv; o.z = az * inv; o.w = aw * inv;
  } else {
    o.x = 0.f; o.y = 0.f; o.z = 0.f; o.w = 0.f;  // empty segment -> zeros
  }
  *(float4*)(out + (size_t)g * HID + lane * 4) = o;
}

extern "C" void kernel_launch(void* const* d_in, const int* in_sizes, int n_in,
                              void* d_out, int out_size, void* d_ws, size_t ws_size,
                              hipStream_t stream) {
  (void)n_in; (void)d_ws; (void)ws_size;
  const float* x   = (const float*)d_in[0];
  const float* w   = (const float*)d_in[1];
  const int*   seg = (const int*)d_in[2];
  float*       out = (float*)d_out;
  const int n_nodes  = in_sizes[2];        // == N_NODES
  const int num_segs = out_size / HID;     // == 8192
  attn_pool_kernel<<<dim3((unsigned)num_segs), dim3(32), 0, stream>>>(
      x, w, seg, out, n_nodes);
}